// MultiheadAttention_85882166051390
// MI455X (gfx1250) — compile-verified
//
#include <hip/hip_runtime.h>
#include <stdint.h>
#include <stddef.h>

// Problem constants (match reference)
#define B_SZ   64
#define T_SZ   1024
#define DK_SZ  64
#define QT     32            // query rows per workgroup
#define SROW   1028          // LDS row stride (floats); 1028 % 64 == 4 -> bank-conflict-free
#define LDS_BYTES (QT * SROW * 4)   // 131,584 B  (< 320 KB WGP LDS)

#define MASK_FILL (-4294967295.0f)  // float(-2**32 + 1)
#define LOG2E     1.4426950408889634f

typedef float v2f __attribute__((ext_vector_type(2)));
typedef float v8f __attribute__((ext_vector_type(8)));

// D = A(16x4) * B(4x16) + C(16x16), fp32, wave32.
// 8-arg family: (neg_a, A, neg_b, B, c_mod, C, reuse_a, reuse_b)
static __device__ __forceinline__ v8f wmma_f32_4(v2f a, v2f b, v8f c) {
  return __builtin_amdgcn_wmma_f32_16x16x4_f32(false, a, false, b, (short)0, c,
                                               false, false);
}

__global__ __launch_bounds__(256) void attn_fwd_kernel(
    const float* __restrict__ Q, const float* __restrict__ K,
    const float* __restrict__ V, const uint8_t* __restrict__ mask,
    const float* __restrict__ qmask, const float* __restrict__ gf,
    float* __restrict__ out_res, float* __restrict__ out_attn) {
  extern __shared__ float S[];  // [QT][SROW] score / attn tile

  const int tid  = threadIdx.x;
  const int wave = tid >> 5;    // 0..7
  const int lane = tid & 31;

  const int bq = blockIdx.x;            // b * (T/QT) + qt
  const int b  = bq >> 5;               // T/QT == 32
  const int qt = bq & 31;
  const int q0 = qt * QT;               // first query row of this block

  const int mrow  = lane & 15;          // M (or N) index within a 16-wide tile
  const int khalf = (lane >> 4) << 1;   // 0 or 2: K-dim sub-offset per A/B layout
  const int rhalf = (lane >> 4) << 3;   // 0 or 8: C/D row offset per layout

  // ---------------- Phase 1: resident Q fragments (A operand) --------------
  // A 16x4 layout: lane l holds rows M=l&15, K = khalf and khalf+1 (2 floats).
  const float* Qb = Q + ((size_t)b * T_SZ + q0) * DK_SZ;
  v2f aQ[2][16];
#pragma unroll
  for (int qs = 0; qs < 2; ++qs) {
#pragma unroll
    for (int dg = 0; dg < 16; ++dg) {
      const float* p = Qb + (size_t)(qs * 16 + mrow) * DK_SZ + dg * 4 + khalf;
      aQ[qs][dg] = *(const v2f*)p;  // global_load_b64, L2-resident
    }
  }

  // ---------------- Phase 2: S = Q K^T via WMMA, into LDS ------------------
  // Wave w owns k-tiles {w, w+8, ..., w+56}. B operand holds K^T, whose
  // 4x16 per-lane layout is identical to A's over a row-major 16x64 tile.
  const float* Kb = K + (size_t)b * T_SZ * DK_SZ;
#pragma unroll 1
  for (int t = 0; t < 8; ++t) {
    const int kt  = wave + t * 8;
    const int kc0 = kt * 16;
    v2f bK[16];
#pragma unroll
    for (int dg = 0; dg < 16; ++dg)
      bK[dg] = *(const v2f*)(Kb + (size_t)(kc0 + mrow) * DK_SZ + dg * 4 + khalf);

#pragma unroll
    for (int qs = 0; qs < 2; ++qs) {
      v8f c = {};
#pragma unroll
      for (int dg = 0; dg < 16; ++dg) c = wmma_f32_4(aQ[qs][dg], bK[dg], c);
      // C layout: VGPR r -> row r + rhalf, col = lane&15 (plus tile offsets)
      const int col = kc0 + mrow;
#pragma unroll
      for (int r = 0; r < 8; ++r)
        S[(qs * 16 + r + rhalf) * SROW + col] = c[r];
    }
  }
  __syncthreads();

  // ---------------- Phase 3: softmax rows in LDS ---------------------------
  // Wave w handles rows 4w..4w+3; 32 lanes sweep 1024 cols, fully coalesced
  // for gf / mask / qmask / attn-out. p[] stays in registers across sweeps.
  for (int i = 0; i < 4; ++i) {
    const int row    = wave * 4 + i;
    const int qrow   = q0 + row;
    const size_t off = ((size_t)b * T_SZ + qrow) * (size_t)T_SZ;

    float p[32];
    float m = -3.402823466e38f;
#pragma unroll
    for (int j = 0; j < 32; ++j) {
      const int c = j * 32 + lane;
      float s = S[row * SROW + c] * 0.125f - gf[off + c];  // /sqrt(64)
      if (mask[off + c]) s = MASK_FILL;
      p[j] = s;
      m = fmaxf(m, s);
    }
#pragma unroll
    for (int o = 16; o; o >>= 1) m = fmaxf(m, __shfl_xor(m, o, 32));

    float sum = 0.0f;
#pragma unroll
    for (int j = 0; j < 32; ++j) {
      p[j] = exp2f((p[j] - m) * LOG2E);  // v_exp_f32
      sum += p[j];
    }
#pragma unroll
    for (int o = 16; o; o >>= 1) sum += __shfl_xor(sum, o, 32);
    const float inv = 1.0f / sum;

#pragma unroll
    for (int j = 0; j < 32; ++j) {
      const int c = j * 32 + lane;
      const float a = p[j] * inv * qmask[off + c];
      S[row * SROW + c] = a;       // for the PV GEMM
      out_attn[off + c] = a;       // second reference output
    }
  }
  __syncthreads();

  // ---------------- Phase 4: O = attn * V via WMMA -------------------------
  // 8 output 16x16 tiles: wave -> (q-subtile, n-tile). A-fragments come from
  // LDS (ds_load_b64, conflict-free with SROW%64==4); V from global (L2).
  const int qs   = wave >> 2;          // 0..1
  const int nt   = wave & 3;           // 0..3
  const int ncol = nt * 16 + mrow;
  const int arow = qs * 16 + mrow;
  const float* Vb = V + (size_t)b * T_SZ * DK_SZ;

  v8f acc = {};
#pragma unroll 4
  for (int kk = 0; kk < T_SZ; kk += 4) {
    const int d0 = kk + khalf;
    v2f a = *(const v2f*)&S[arow * SROW + d0];
    v2f bb;
    bb.x = Vb[(size_t)d0 * DK_SZ + ncol];
    bb.y = Vb[(size_t)(d0 + 1) * DK_SZ + ncol];
    acc = wmma_f32_4(a, bb, acc);
  }

#pragma unroll
  for (int r = 0; r < 8; ++r) {
    const int row = q0 + qs * 16 + r + rhalf;
    out_res[((size_t)b * T_SZ + row) * DK_SZ + ncol] = acc[r];
  }
}

extern "C" void kernel_launch(void* const* d_in, const int* in_sizes, int n_in,
                              void* d_out, int out_size, void* d_ws,
                              size_t ws_size, hipStream_t stream) {
  // setup_inputs() order: key, value, query, mask(bool), query_mask, gaussian_factor
  const float*   K    = (const float*)d_in[0];
  const float*   V    = (const float*)d_in[1];
  const float*   Q    = (const float*)d_in[2];
  const uint8_t* mask = (const uint8_t*)d_in[3];  // jnp.bool_ -> 1 byte/elem
  const float*   qm   = (const float*)d_in[4];
  const float*   gf   = (const float*)d_in[5];

  float* out_res  = (float*)d_out;                       // (B,T,DK) first
  float* out_attn = out_res + (size_t)B_SZ * T_SZ * DK_SZ;  // then (B,T,T)

  const int grid = B_SZ * (T_SZ / QT);  // 2048 workgroups
  attn_fwd_kernel<<<grid, 256, LDS_BYTES, stream>>>(Q, K, V, mask, qm, gf,
                                                    out_res, out_attn);
}